// Baseline_SelfGCN_12051678232604
// MI455X (gfx1250) — compile-verified
//
#include <hip/hip_runtime.h>
#include <cstddef>

typedef float v2f __attribute__((ext_vector_type(2)));
typedef float v8f __attribute__((ext_vector_type(8)));

#define BD 64
#define CD 2048
#define NPD 9
#define MTOT 1152            // 2 passes * 64 b * 9 parts
#define OUTW 43008           // per-batch output width
#define EPSBN 1e-5f

// ---------------------------------------------------------------------------
// Kernel 1: per-batch mask histogram -> drop label + 1/(count+1e-8)
// ---------------------------------------------------------------------------
__global__ void k_maskprep(const int* __restrict__ mask,
                           int* __restrict__ dropBuf,
                           float* __restrict__ invBuf) {
  __shared__ int cnt[16];
  int b = blockIdx.x;
  int tid = threadIdx.x;
  if (tid < 16) cnt[tid] = 0;
  __syncthreads();
  // mr[i][j] = mask[b,0,16*i,16*j] ; i=tid>>4, j=tid&15
  int i = tid >> 4, j = tid & 15;
  int v = mask[(size_t)b * 65536 + (size_t)i * 4096 + j * 16];
  atomicAdd(&cnt[v], 1);
  __syncthreads();
  if (tid == 0) {
    int drop = 1;  // argmax over all-false -> index 0 -> drop = 1
    for (int p = 0; p < NPD; ++p) {
      if (cnt[p + 1] > 0) { drop = p + 1; break; }
    }
    dropBuf[b] = drop;
  }
  if (tid < NPD) {
    invBuf[b * NPD + tid] = 1.0f / ((float)cnt[tid + 1] + 1e-8f);
  }
}

// ---------------------------------------------------------------------------
// Kernel 2: feat_global = mean over HW, then BN -> out[b][0:2048]
// ---------------------------------------------------------------------------
__global__ void k_globalmean(const float* __restrict__ xg,
                             const float* __restrict__ g,
                             const float* __restrict__ be,
                             const float* __restrict__ rm,
                             const float* __restrict__ rv,
                             float* __restrict__ out) {
  int b = blockIdx.x;
  int c = blockIdx.y * 256 + threadIdx.x;
  const float4* p = reinterpret_cast<const float4*>(
      xg + ((size_t)b * CD + c) * 256);
  float s = 0.f;
#pragma unroll 4
  for (int t = 0; t < 64; ++t) {
    float4 v = p[t];
    s += v.x + v.y + v.z + v.w;
  }
  float mean = s * (1.0f / 256.0f);
  float val = (mean - rm[c]) * (1.0f / sqrtf(rv[c] + EPSBN)) * g[c] + be[c];
  out[(size_t)b * OUTW + c] = val;
}

// ---------------------------------------------------------------------------
// Kernel 3: masked pooling.  X rows [0..575]=mask_feat, [576..1151]=self_feat
// Per-thread private LDS accumulator acc[tid*10 + label]; label is
// block-uniform per pixel -> conflict-free banking.
// ---------------------------------------------------------------------------
__global__ void k_pool(const float* __restrict__ xg,
                       const int* __restrict__ mask,
                       const int* __restrict__ dropBuf,
                       const float* __restrict__ invBuf,
                       float* __restrict__ X) {
  __shared__ float acc[256 * 10];
  __shared__ int labels[256];
  int b = blockIdx.x;
  int tid = threadIdx.x;
  int c = blockIdx.y * 256 + tid;
  labels[tid] = mask[(size_t)b * 65536 + (size_t)(tid >> 4) * 4096 + (tid & 15) * 16];
#pragma unroll
  for (int p = 0; p < 10; ++p) acc[tid * 10 + p] = 0.f;
  __syncthreads();

  const float4* src = reinterpret_cast<const float4*>(
      xg + ((size_t)b * CD + c) * 256);
  for (int j = 0; j < 64; ++j) {
    float4 v = src[j];
    int base = j * 4;
    acc[tid * 10 + labels[base + 0]] += v.x;
    acc[tid * 10 + labels[base + 1]] += v.y;
    acc[tid * 10 + labels[base + 2]] += v.z;
    acc[tid * 10 + labels[base + 3]] += v.w;
  }
  int drop = dropBuf[b];
#pragma unroll
  for (int p = 0; p < NPD; ++p) {
    float val = acc[tid * 10 + p + 1] * invBuf[b * NPD + p];
    X[((size_t)(b * NPD + p)) * CD + c] = val;
    X[((size_t)(576 + b * NPD + p)) * CD + c] = (p + 1 == drop) ? 0.f : val;
  }
}

// ---------------------------------------------------------------------------
// Kernel 4: S = X @ W  using V_WMMA_F32_16X16X4_F32.
// M=1152, N=K=2048, row-major. One wave -> 32x64 strip (2 M-tiles x 4 N-tiles
// = 8 WMMA tiles). B fragments shared by both M-tiles: per k-step
// 2xb64 (A) + 8xb32 (B) = 10 VMEM for 8 WMMAs (1.25 vmem/wmma).
// 144 blocks * 8 waves = 1152 jobs = 36 Mtiles * 32 Nsupertiles (exact cover).
// ---------------------------------------------------------------------------
__global__ void __launch_bounds__(256) k_gemm(const float* __restrict__ X,
                                              const float* __restrict__ W,
                                              float* __restrict__ S) {
  const int K = CD, N = CD;
  int tid = threadIdx.x;
  int lane = tid & 31;
  int wave = tid >> 5;
  int job = blockIdx.x * 8 + wave;
  int tm = job >> 5;       // 0..35
  int tn = job & 31;       // 0..31
  int m0 = tm * 32;
  int n0 = tn * 64;
  int lm = lane & 15;
  int hi = lane >> 4;      // 0 or 1

  // A frag: lane<16 holds (K=k,k+1) of row m0+lm ; lane>=16 holds (K=k+2,k+3)
  const float* Aptr0 = X + (size_t)(m0 + lm) * K + 2 * hi;
  const float* Aptr1 = X + (size_t)(m0 + 16 + lm) * K + 2 * hi;
  // B frag: lane<16 holds W[k][n+lm], W[k+1][n+lm]; lane>=16: rows k+2,k+3
  const float* Bptr = W + (size_t)(2 * hi) * N + n0 + lm;

  v8f accA[4];
  v8f accB[4];
#pragma unroll
  for (int t = 0; t < 4; ++t) { accA[t] = (v8f)(0.0f); accB[t] = (v8f)(0.0f); }

  for (int k = 0; k < K; k += 4) {
    v2f a0 = *reinterpret_cast<const v2f*>(Aptr0 + k);
    v2f a1 = *reinterpret_cast<const v2f*>(Aptr1 + k);
    const float* bk = Bptr + (size_t)k * N;
    v2f bf[4];
#pragma unroll
    for (int t = 0; t < 4; ++t) {
      bf[t].x = bk[t * 16];
      bf[t].y = bk[(size_t)N + t * 16];
    }
#pragma unroll
    for (int t = 0; t < 4; ++t) {
      accA[t] = __builtin_amdgcn_wmma_f32_16x16x4_f32(
          false, a0, false, bf[t], (short)0, accA[t], false, false);
      accB[t] = __builtin_amdgcn_wmma_f32_16x16x4_f32(
          false, a1, false, bf[t], (short)0, accB[t], false, false);
    }
  }

  // C/D layout: VGPR r -> row r (lanes 0-15), row r+8 (lanes 16-31); N = lane%16
#pragma unroll
  for (int t = 0; t < 4; ++t) {
    float* dst0 = S + (size_t)(m0 + 8 * hi) * N + n0 + t * 16 + lm;
    float* dst1 = S + (size_t)(m0 + 16 + 8 * hi) * N + n0 + t * 16 + lm;
#pragma unroll
    for (int r = 0; r < 8; ++r) {
      dst0[(size_t)r * N] = accA[t][r];
      dst1[(size_t)r * N] = accB[t][r];
    }
  }
}

// ---------------------------------------------------------------------------
// Kernel 5: o = adj @ s + bias ; BN over flattened (p*C+d) ; ReLU.
// grid (64, 16): half = blockIdx.y>>3, cgroup = blockIdx.y&7.
// final==0 -> write to xout rows;  final==1 -> write to d_out cat regions.
// ---------------------------------------------------------------------------
__global__ void k_combine(const float* __restrict__ S,
                          const float* __restrict__ adj,
                          const float* __restrict__ bias,
                          const float* __restrict__ g,
                          const float* __restrict__ be,
                          const float* __restrict__ rm,
                          const float* __restrict__ rv,
                          float* __restrict__ xout,
                          float* __restrict__ finalout,
                          int final_mode) {
  __shared__ float sadj[NPD * NPD];
  int b = blockIdx.x;
  int half = blockIdx.y >> 3;
  int cg = blockIdx.y & 7;
  int tid = threadIdx.x;
  int d = cg * 256 + tid;
  if (tid < NPD * NPD) sadj[tid] = adj[b * NPD * NPD + tid];
  __syncthreads();

  float sq[NPD];
  size_t rbase = (size_t)(half * 576 + b * NPD) * CD + d;
#pragma unroll
  for (int q = 0; q < NPD; ++q) sq[q] = S[rbase + (size_t)q * CD];

  float bv = bias[d];
#pragma unroll
  for (int p = 0; p < NPD; ++p) {
    float o = bv;
#pragma unroll
    for (int q = 0; q < NPD; ++q) o += sadj[p * NPD + q] * sq[q];
    int idx = p * CD + d;
    float val = (o - rm[idx]) * (1.0f / sqrtf(rv[idx] + EPSBN)) * g[idx] + be[idx];
    val = val > 0.f ? val : 0.f;
    if (final_mode) {
      finalout[(size_t)b * OUTW + 6144 + half * 18432 + p * CD + d] = val;
    } else {
      xout[(size_t)(half * 576 + b * NPD + p) * CD + d] = val;
    }
  }
}

// ---------------------------------------------------------------------------
// Kernel 6: feat_gcn/feat_self = mean over p of x2 (read back from cat
// regions of out), then BN with gn params -> out[b][2048:4096], [4096:6144]
// ---------------------------------------------------------------------------
__global__ void k_finmean(const float* __restrict__ g,
                          const float* __restrict__ be,
                          const float* __restrict__ rm,
                          const float* __restrict__ rv,
                          float* __restrict__ out) {
  int b = blockIdx.x;
  int d = blockIdx.y * 256 + threadIdx.x;
  size_t base = (size_t)b * OUTW;
  float sg = 0.f, ss = 0.f;
#pragma unroll
  for (int p = 0; p < NPD; ++p) {
    sg += out[base + 6144 + p * CD + d];
    ss += out[base + 24576 + p * CD + d];
  }
  float inv9 = 1.0f / 9.0f;
  float isd = (1.0f / sqrtf(rv[d] + EPSBN)) * g[d];
  out[base + 2048 + d] = (sg * inv9 - rm[d]) * isd + be[d];
  out[base + 4096 + d] = (ss * inv9 - rm[d]) * isd + be[d];
}

// ---------------------------------------------------------------------------
extern "C" void kernel_launch(void* const* d_in, const int* in_sizes, int n_in,
                              void* d_out, int out_size, void* d_ws, size_t ws_size,
                              hipStream_t stream) {
  const float* x_global = (const float*)d_in[0];
  const float* x_gcn    = (const float*)d_in[1];
  const int*   mask     = (const int*)d_in[2];
  const float* adj      = (const float*)d_in[3];
  const float* W1  = (const float*)d_in[4];
  const float* b1  = (const float*)d_in[5];
  const float* g1  = (const float*)d_in[6];
  const float* be1 = (const float*)d_in[7];
  const float* rm1 = (const float*)d_in[8];
  const float* rv1 = (const float*)d_in[9];
  const float* W2  = (const float*)d_in[10];
  const float* b2  = (const float*)d_in[11];
  const float* g2  = (const float*)d_in[12];
  const float* be2 = (const float*)d_in[13];
  const float* rm2 = (const float*)d_in[14];
  const float* rv2 = (const float*)d_in[15];
  const float* gb_g  = (const float*)d_in[16];
  const float* gb_b  = (const float*)d_in[17];
  const float* gb_rm = (const float*)d_in[18];
  const float* gb_rv = (const float*)d_in[19];
  const float* gn_g  = (const float*)d_in[20];
  const float* gn_b  = (const float*)d_in[21];
  const float* gn_rm = (const float*)d_in[22];
  const float* gn_rv = (const float*)d_in[23];
  float* out = (float*)d_out;

  // workspace layout (floats): [0..63]=drop(int), [64..639]=inv_denom,
  // X at 1024 (1152*2048), S at 1024+2359296 (1152*2048)
  int*   dropBuf = (int*)d_ws;
  float* invBuf  = (float*)d_ws + 64;
  float* Xbuf    = (float*)d_ws + 1024;
  float* Sbuf    = Xbuf + (size_t)MTOT * CD;

  k_maskprep<<<BD, 256, 0, stream>>>(mask, dropBuf, invBuf);
  k_globalmean<<<dim3(BD, 8), 256, 0, stream>>>(x_global, gb_g, gb_b, gb_rm, gb_rv, out);
  k_pool<<<dim3(BD, 8), 256, 0, stream>>>(x_gcn, mask, dropBuf, invBuf, Xbuf);

  // layer 1: S = X @ W1 ; x1 = relu(bn(adj@S + b1)) -> reuse Xbuf
  k_gemm<<<144, 256, 0, stream>>>(Xbuf, W1, Sbuf);
  k_combine<<<dim3(BD, 16), 256, 0, stream>>>(Sbuf, adj, b1, g1, be1, rm1, rv1,
                                              Xbuf, nullptr, 0);
  // layer 2: S2 = x1 @ W2 ; x2 = relu(bn(adj@S2 + b2)) -> straight to out cats
  k_gemm<<<144, 256, 0, stream>>>(Xbuf, W2, Sbuf);
  k_combine<<<dim3(BD, 16), 256, 0, stream>>>(Sbuf, adj, b2, g2, be2, rm2, rv2,
                                              nullptr, out, 1);
  // means over parts + BN(gn)
  k_finmean<<<dim3(BD, 8), 256, 0, stream>>>(gn_g, gn_b, gn_rm, gn_rv, out);
}